// GroupNorm_18743237280237
// MI455X (gfx1250) — compile-verified
//
#include <hip/hip_runtime.h>

// ---------------------------------------------------------------------------
// GroupNorm over irreps RS=[(64,1),(32,3),(16,5),(8,7)], x[4,296,32,32,32] f32
// Memory-bound: stats pass (read 155MB) + apply pass (read+write; read should
// hit the 192MB L2 warmed by pass 1). V_WMMA_F32_16X16X4_F32 with B=ones is
// the wave32 reduction engine: 32 lane partials -> 16 row sums in one op.
// ---------------------------------------------------------------------------

typedef float v2f __attribute__((ext_vector_type(2)));
typedef float v8f __attribute__((ext_vector_type(8)));

#define NCH   296
#define NB    4
#define S_CH  32768          // 32*32*32 spatial per channel
#define S4_CH 8192           // float4 per channel

// Wave32 half-reduction via V_WMMA_F32_16X16X4_F32.
// A layout (16x4 f32): lanes 0-15 hold (row=lane, K=0/1) in vgpr0/1,
// lanes 16-31 hold (row=lane-16, K=2/3). Set a0=p, a1=0, B=ones:
//   D[i][j] = p[i] + p[i+16] = q[i]
// D layout: vgpr r = q[r] for lanes<16, q[8+r] for lanes>=16. Summing the 8
// D vgprs gives: lanes<16 -> sum q[0..7], lanes>=16 -> sum q[8..15].
// Wave total = (lane0 value) + (lane16 value); both written to LDS by caller.
__device__ __forceinline__ float wave_halfreduce_wmma(float p) {
    v2f a; a[0] = p;    a[1] = 0.0f;
    v2f b; b[0] = 1.0f; b[1] = 1.0f;
    v8f c = {};
    v8f d = __builtin_amdgcn_wmma_f32_16x16x4_f32(
        /*neg_a=*/false, a, /*neg_b=*/false, b,
        /*c_mod=*/(short)0, c, /*reuse_a=*/false, /*reuse_b=*/false);
    return ((d[0] + d[1]) + (d[2] + d[3])) + ((d[4] + d[5]) + (d[6] + d[7]));
}

// ---------------------------------------------------------------------------
// Kernel 1: one block per (b,c); reduce 32768 elems -> sumsq (+ sum for c<64)
// Multiple independent accumulators: no VALU dependency stalls, HBM-limited.
// ---------------------------------------------------------------------------
__global__ __launch_bounds__(256) void gn_stats(const float* __restrict__ x,
                                                float* __restrict__ ws_sumsq,
                                                float* __restrict__ ws_sum) {
    const int bc = blockIdx.x;            // b*296 + c
    const int c  = bc % NCH;
    const int t  = threadIdx.x;
    const float4* __restrict__ p =
        reinterpret_cast<const float4*>(x) + (size_t)bc * S4_CH;

    float sq0 = 0.f, sq1 = 0.f, sq2 = 0.f, sq3 = 0.f;
    float sm0 = 0.f, sm1 = 0.f;
    #pragma unroll 8
    for (int i = 0; i < 32; ++i) {
        float4 v = p[i * 256 + t];
        sq0 = fmaf(v.x, v.x, sq0);
        sq1 = fmaf(v.y, v.y, sq1);
        sq2 = fmaf(v.z, v.z, sq2);
        sq3 = fmaf(v.w, v.w, sq3);
        sm0 += v.x + v.y;
        sm1 += v.z + v.w;
    }
    const float psq  = (sq0 + sq1) + (sq2 + sq3);
    const float psum = sm0 + sm1;

    __shared__ float lds_sq[16];          // 8 waves x 2 half-sums
    __shared__ float lds_sum[16];
    const bool needSum = (c < 64);        // uniform across block
    const bool halfLead = ((t & 15) == 0);   // lane 0 and lane 16 of each wave
    const int  slot = t >> 4;                // 0..15

    float hsq = wave_halfreduce_wmma(psq);   // EXEC all-1s here
    if (halfLead) lds_sq[slot] = hsq;
    if (needSum) {
        float hsm = wave_halfreduce_wmma(psum);
        if (halfLead) lds_sum[slot] = hsm;
    }
    __syncthreads();

    if (t == 0) {
        float s = 0.0f;
        #pragma unroll
        for (int i = 0; i < 16; ++i) s += lds_sq[i];
        ws_sumsq[bc] = s;
        if (needSum) {
            float s2 = 0.0f;
            #pragma unroll
            for (int i = 0; i < 16; ++i) s2 += lds_sum[i];
            ws_sum[(bc / NCH) * 64 + c] = s2;
        }
    }
}

// ---------------------------------------------------------------------------
// Kernel 2: per batch, fold channel partials into per-(b,c) fused {A,B} pairs
//   out = x * A[b,c] + B[b,c]
// group0 (d==1): A = invstd0*w[c], B = bias[c] - mean*A  (centering folded in)
// ---------------------------------------------------------------------------
__global__ __launch_bounds__(320) void gn_coef(const float* __restrict__ ws_sumsq,
                                               const float* __restrict__ ws_sum,
                                               const float* __restrict__ weight,
                                               const float* __restrict__ bias,
                                               float2* __restrict__ coef) {
    const int b = blockIdx.x;
    const int t = threadIdx.x;
    __shared__ float ss[NCH];
    __shared__ float inv[4];
    __shared__ float mean0;

    if (t < NCH) ss[t] = ws_sumsq[b * NCH + t];
    __syncthreads();

    if (t == 0) {
        float g0 = 0.f, g1 = 0.f, g2 = 0.f, g3 = 0.f, sm = 0.f;
        for (int i = 0;   i < 64;  ++i) g0 += ss[i];
        for (int i = 64;  i < 160; ++i) g1 += ss[i];
        for (int i = 160; i < 240; ++i) g2 += ss[i];
        for (int i = 240; i < 296; ++i) g3 += ss[i];
        for (int i = 0;   i < 64;  ++i) sm += ws_sum[b * 64 + i];
        const float S = (float)S_CH;
        const float n0 = 64.f * S, n1 = 32.f * S, n2 = 16.f * S, n3 = 8.f * S;
        const float mn = sm / n0;
        mean0  = mn;
        inv[0] = 1.0f / sqrtf(g0 / n0 - mn * mn + 1e-5f);  // E[x^2]-mean^2
        inv[1] = 1.0f / sqrtf(g1 / n1 + 1e-5f);            // /= m*S (not m*d*S)
        inv[2] = 1.0f / sqrtf(g2 / n2 + 1e-5f);
        inv[3] = 1.0f / sqrtf(g3 / n3 + 1e-5f);
    }
    __syncthreads();

    if (t < NCH) {
        float A, Bc = 0.0f;
        if (t < 64) {
            A  = inv[0] * weight[t];
            Bc = bias[t] - mean0 * A;
        } else if (t < 160) {
            A = inv[1] * weight[64  + (t - 64)  / 3];
        } else if (t < 240) {
            A = inv[2] * weight[96  + (t - 160) / 5];
        } else {
            A = inv[3] * weight[112 + (t - 240) / 7];
        }
        coef[b * NCH + t] = make_float2(A, Bc);
    }
}

// ---------------------------------------------------------------------------
// Kernel 3: elementwise out = x*A + B, float4 loads/stores, grid-stride
// (exactly 8 float4 per thread at 4736x256; x read should be L2-resident)
// ---------------------------------------------------------------------------
__global__ __launch_bounds__(256) void gn_apply(const float* __restrict__ x,
                                                const float2* __restrict__ coef,
                                                float* __restrict__ out,
                                                int n4) {
    const float4* __restrict__ xv = reinterpret_cast<const float4*>(x);
    float4* __restrict__ ov = reinterpret_cast<float4*>(out);
    const int stride = gridDim.x * blockDim.x;
    for (int v = blockIdx.x * blockDim.x + threadIdx.x; v < n4; v += stride) {
        const int bc = v >> 13;                 // v / 8192 = b*296 + c
        const float2 ab = coef[bc];             // single b64 load
        float4 d = xv[v];
        d.x = fmaf(d.x, ab.x, ab.y);
        d.y = fmaf(d.y, ab.x, ab.y);
        d.z = fmaf(d.z, ab.x, ab.y);
        d.w = fmaf(d.w, ab.x, ab.y);
        ov[v] = d;
    }
}

// ---------------------------------------------------------------------------
extern "C" void kernel_launch(void* const* d_in, const int* in_sizes, int n_in,
                              void* d_out, int out_size, void* d_ws, size_t ws_size,
                              hipStream_t stream) {
    const float* x      = (const float*)d_in[0];   // [4,296,32,32,32]
    const float* weight = (const float*)d_in[1];   // [120]
    const float* bias   = (const float*)d_in[2];   // [64]
    float* out = (float*)d_out;

    // workspace layout (floats): sumsq[1184] | sum[256] | coef[2*1184]
    float*  w        = (float*)d_ws;
    float*  ws_sumsq = w;
    float*  ws_sum   = w + NB * NCH;               // +1184
    float2* coef     = (float2*)(ws_sum + NB * 64);// +256, 8-byte aligned

    gn_stats<<<NB * NCH, 256, 0, stream>>>(x, ws_sumsq, ws_sum);
    gn_coef<<<NB, 320, 0, stream>>>(ws_sumsq, ws_sum, weight, bias, coef);

    const int n4 = NB * NCH * S4_CH;               // 9,699,328 float4
    gn_apply<<<4736, 256, 0, stream>>>(x, coef, out, n4);
}